// SSF_Extractor_2362232013380
// MI455X (gfx1250) — compile-verified
//
#include <hip/hip_runtime.h>
#include <stdint.h>

// ---------------------------------------------------------------------------
// Fused 5x5 local-variance -> bilinear(align_corners) upsample -> multiply.
// x: [8,64,256,256] f32.  One workgroup (256 thr, 8 waves) = one 64x64 output
// tile of one image plane.  Input patch is DMA'd to LDS with the CDNA5 async
// global->LDS path (ASYNCcnt), separable running box-sums build the variance
// grid in LDS, then bilinear + multiply with coalesced non-temporal b128
// stores.  ~73 KB LDS/block -> up to 4 blocks (32 waves) per 320 KB WGP.
// ---------------------------------------------------------------------------

typedef float vf4 __attribute__((ext_vector_type(4)));   // native clang vector

namespace {
constexpr int IMG_H   = 256;
constexpr int IMG_W   = 256;
constexpr int NPROWS  = 69;   // patch rows   (<= 65 grid rows + 4)
constexpr int LDSW    = 72;   // patch cols / LDS row stride (18 x b128)
constexpr int NGRID   = 65;   // grid rows/cols covered per tile (max)
constexpr int VARW    = 66;   // variance LDS row stride
constexpr int NTHREADS = 256;
}

__global__ __launch_bounds__(NTHREADS)
void ssf_var_fused_kernel(const float* __restrict__ x,
                          float* __restrict__ out)
{
    __shared__ __align__(16) float xp[NPROWS * LDSW];   // input patch
    __shared__ float V1s[NGRID * LDSW];                 // vertical 5-sum of x
    __shared__ float V2s[NGRID * LDSW];                 // vertical 5-sum of x^2
    __shared__ float VARs[NGRID * VARW];                // variance grid

    const int tid  = threadIdx.x;
    const int bid  = blockIdx.x;
    const int img  = bid >> 4;            // 512 image planes
    const int tile = bid & 15;
    const int oy   = (tile >> 2) << 6;    // tile origin in output
    const int ox   = (tile & 3)  << 6;

    // First grid (variance) row/col touched by this tile: floor(o*251/255).
    const int gy0 = (oy * 251) / 255;
    const int gx0 = (ox * 251) / 255;
    const int cb  = gx0 & ~3;             // 16B-aligned patch column base
    const int dxo = gx0 - cb;             // 0..3 column phase

    const float* __restrict__ src = x + (size_t)img * (IMG_H * IMG_W);

    // ---- Stage 1: async-copy 69x72 patch (rows gy0.., cols cb..) to LDS ----
    for (int k = tid; k < NPROWS * (LDSW / 4); k += NTHREADS) {
        int pr = k / (LDSW / 4);
        int pc = (k - pr * (LDSW / 4)) * 4;
        int gr = gy0 + pr; gr = gr > (IMG_H - 1) ? (IMG_H - 1) : gr;  // edge clamp
        int gc = cb  + pc; gc = gc > (IMG_W - 4) ? (IMG_W - 4) : gc;  // clamped cells unused
        const float* gp  = src + gr * IMG_W + gc;
        uint32_t lofs = (uint32_t)(uintptr_t)(&xp[pr * LDSW + pc]);
#if defined(__gfx1250__)
        asm volatile("global_load_async_to_lds_b128 %0, %1, off"
                     :: "v"(lofs), "v"(gp)
                     : "memory");
#else
        (void)lofs;
        *(vf4*)&xp[pr * LDSW + pc] = *(const vf4*)gp;
#endif
    }
#if defined(__gfx1250__)
    asm volatile("s_wait_asynccnt 0x0" ::: "memory");
#endif
    __syncthreads();

    // ---- Stage 2: vertical running 5-sums of x and x^2 ---------------------
    // 72 columns x 5 strips of 13 grid rows (5*13 = 65).
    for (int t = tid; t < LDSW * 5; t += NTHREADS) {
        int c  = t / 5;
        int g  = t - c * 5;
        int r0 = g * 13;
        float s = 0.0f, s2 = 0.0f;
#pragma unroll
        for (int k = 0; k < 5; ++k) {
            float v = xp[(r0 + k) * LDSW + c];
            s += v; s2 += v * v;
        }
        V1s[r0 * LDSW + c] = s;
        V2s[r0 * LDSW + c] = s2;
        for (int r = r0 + 1; r < r0 + 13; ++r) {
            float vo = xp[(r - 1) * LDSW + c];
            float vi = xp[(r + 4) * LDSW + c];
            s  += vi - vo;
            s2 += vi * vi - vo * vo;
            V1s[r * LDSW + c] = s;
            V2s[r * LDSW + c] = s2;
        }
    }
    __syncthreads();

    // ---- Stage 3: horizontal running 5-sums -> variance grid ---------------
    // 65 rows x 5 strips of 13 grid cols.  var = (S2 - S*S/25) / 24.
    const float invN   = 1.0f / 25.0f;
    const float invNm1 = 1.0f / 24.0f;
    for (int t = tid; t < NGRID * 5; t += NTHREADS) {
        int r  = t / 5;
        int g  = t - r * 5;
        int c0 = g * 13;
        const float* v1r = &V1s[r * LDSW];
        const float* v2r = &V2s[r * LDSW];
        float s = 0.0f, s2 = 0.0f;
#pragma unroll
        for (int k = 0; k < 5; ++k) {
            s  += v1r[c0 + dxo + k];
            s2 += v2r[c0 + dxo + k];
        }
        VARs[r * VARW + c0] = (s2 - s * s * invN) * invNm1;
        for (int c = c0 + 1; c < c0 + 13; ++c) {
            s  += v1r[c + dxo + 4] - v1r[c + dxo - 1];
            s2 += v2r[c + dxo + 4] - v2r[c + dxo - 1];
            VARs[r * VARW + c] = (s2 - s * s * invN) * invNm1;
        }
    }
    __syncthreads();

    // ---- Stage 4: bilinear resample (align_corners) * x --------------------
    // Thread -> fixed 4-column group; iterate 4 rows.  Column interpolation
    // state (exact integer floor + fractional weight) computed once.
    float* __restrict__ dst = out + (size_t)img * (IMG_H * IMG_W);
    const float inv255 = 1.0f / 255.0f;

    const int lc4 = (tid & 15) << 2;        // local output col group (x4)
    const int lr0 = tid >> 4;               // base local row 0..15

    int   lx0[4], lx1[4];
    float wx[4];
#pragma unroll
    for (int j = 0; j < 4; ++j) {
        int xg   = ox + lc4 + j;
        int xnum = xg * 251;
        int x0g  = xnum / 255;                          // exact floor
        wx[j]    = (float)(xnum - x0g * 255) * inv255;  // exact fractional
        lx0[j]   = x0g - gx0;
        lx1[j]   = (x0g < 251) ? (lx0[j] + 1) : lx0[j]; // clamp to last col
    }

#pragma unroll
    for (int it = 0; it < 4; ++it) {
        int lr   = lr0 + it * 16;          // local output row 0..63
        int y    = oy + lr;
        int ynum = y * 251;
        int y0g  = ynum / 255;
        float wy = (float)(ynum - y0g * 255) * inv255;
        int ly0  = y0g - gy0;
        int ly1  = (y0g < 251) ? (ly0 + 1) : ly0;      // clamp to last row
        const float* vr0  = &VARs[ly0 * VARW];
        const float* vr1  = &VARs[ly1 * VARW];
        const float* xrow = &xp[(y - gy0) * LDSW];

        vf4 res;
#pragma unroll
        for (int j = 0; j < 4; ++j) {
            float a = vr0[lx0[j]], b = vr0[lx1[j]];
            float c = vr1[lx0[j]], d = vr1[lx1[j]];
            float top = a + (b - a) * wx[j];
            float bot = c + (d - c) * wx[j];
            float v   = top + (bot - top) * wy;
            res[j] = xrow[ox + lc4 + j - cb] * v;
        }
        // Output is write-once / never re-read: non-temporal store keeps the
        // 134 MB output stream from evicting the reusable input halos in L2.
        __builtin_nontemporal_store(res, (vf4*)(&dst[(size_t)y * IMG_W + ox + lc4]));
    }
}

extern "C" void kernel_launch(void* const* d_in, const int* in_sizes, int n_in,
                              void* d_out, int out_size, void* d_ws, size_t ws_size,
                              hipStream_t stream) {
    (void)n_in; (void)d_ws; (void)ws_size; (void)out_size;
    const float* x = (const float*)d_in[0];
    float* out = (float*)d_out;
    const int nimg   = in_sizes[0] / (IMG_H * IMG_W);  // 8*64 = 512 planes
    const int blocks = nimg * 16;                      // 16 tiles per plane
    ssf_var_fused_kernel<<<blocks, NTHREADS, 0, stream>>>(x, out);
}